// EfficientDetONNX_66718021976315
// MI455X (gfx1250) — compile-verified
//
#include <hip/hip_runtime.h>
#include <hip/hip_bf16.h>
#include <stdint.h>

#define DET_THRESHOLD 0.05f
#define IOU_THRESHOLD 0.5f
#define MAX_DET       100
#define IMG_W         512.0f
#define IMG_H         512.0f

typedef unsigned int v4u __attribute__((ext_vector_type(4)));
typedef int          v8i __attribute__((ext_vector_type(8)));
typedef int          v4i __attribute__((ext_vector_type(4)));

#if defined(__HIP_DEVICE_COMPILE__) && defined(__gfx1250__) && \
    __has_builtin(__builtin_amdgcn_tensor_load_to_lds) &&       \
    __has_builtin(__builtin_amdgcn_s_wait_tensorcnt)
#define USE_TDM 1
#else
#define USE_TDM 0
#endif

// ---------------------------------------------------------------------------
// Kernel 1: per-anchor box decode + class max + sigmoid score + packed sort key
// key = (orderable_score_bits << 32) | (0xFFFFFFFF - anchor_idx)
//   -> sorting keys descending == stable argsort(-score) of the reference
//   -> invalid (score <= 0.05) encoded as key == 0 (sorts last, detectable)
// ---------------------------------------------------------------------------
__global__ void decode_score_kernel(const float* __restrict__ reg,
                                    const float* __restrict__ cls,
                                    const float* __restrict__ anch,
                                    unsigned long long* __restrict__ wsKeys,
                                    float* __restrict__ wsBoxes,
                                    float* __restrict__ wsScores,
                                    float* __restrict__ wsCls,
                                    int N, int C)
{
    int n = blockIdx.x * blockDim.x + threadIdx.x;
    if (n >= N) return;

    // Warm L2 for the next tile's classification rows (global_prefetch_b8).
    if (n + blockDim.x < N)
        __builtin_prefetch(cls + (size_t)(n + blockDim.x) * (size_t)C, 0, 3);

    float a0 = anch[n * 4 + 0], a1 = anch[n * 4 + 1];
    float a2 = anch[n * 4 + 2], a3 = anch[n * 4 + 3];
    float d0 = reg[n * 4 + 0], d1 = reg[n * 4 + 1];
    float d2 = reg[n * 4 + 2], d3 = reg[n * 4 + 3];

    float w  = a2 - a0, h = a3 - a1;
    float cx = a0 + 0.5f * w, cy = a1 + 0.5f * h;
    float pcx = cx + d0 * 0.1f * w;
    float pcy = cy + d1 * 0.1f * h;
    float pw  = expf(d2 * 0.2f) * w;
    float ph  = expf(d3 * 0.2f) * h;

    float x1 = fminf(fmaxf(pcx - 0.5f * pw, 0.0f), IMG_W);
    float y1 = fminf(fmaxf(pcy - 0.5f * ph, 0.0f), IMG_H);
    float x2 = fminf(fmaxf(pcx + 0.5f * pw, 0.0f), IMG_W);
    float y2 = fminf(fmaxf(pcy + 0.5f * ph, 0.0f), IMG_H);

    // sigmoid is monotone: max/argmax over logits == max/argmax over sigmoid
    const float* cr = cls + (size_t)n * (size_t)C;
    float best = cr[0];
    int   bc   = 0;
    for (int c = 1; c < C; ++c) {
        float v = cr[c];
        if (v > best) { best = v; bc = c; }   // strict '>' keeps first max (argmax)
    }
    float score = 1.0f / (1.0f + expf(-best));

    unsigned long long key = 0ull;
    if (score > DET_THRESHOLD) {
        unsigned int u = __float_as_uint(score) | 0x80000000u;  // positive floats
        key = ((unsigned long long)u << 32) |
              (unsigned long long)(0xFFFFFFFFu - (unsigned)n);
    }
    wsKeys[n]          = key;
    wsBoxes[n * 4 + 0] = x1;
    wsBoxes[n * 4 + 1] = y1;
    wsBoxes[n * 4 + 2] = x2;
    wsBoxes[n * 4 + 3] = y2;
    wsScores[n]        = score;
    wsCls[n]           = (float)bc;
}

#if USE_TDM
// Issue one TDM 1D load: nElem8 x 8-byte elements, global -> LDS.
__device__ __forceinline__ void tdm_load_1d(unsigned int lds_addr,
                                            const void* gptr,
                                            unsigned int nElem8)
{
    unsigned long long gaddr = (unsigned long long)(uintptr_t)gptr;
    unsigned long long td0   = (unsigned long long)nElem8;

    union { unsigned long long q[2]; v4u v; } g0;
    g0.q[0] = 1ull /*count=1*/ | ((unsigned long long)lds_addr << 32);
    g0.q[1] = (gaddr & 0x01FFFFFFFFFFFFFFull) | (2ull << 62);   // type=2 "image"

    union { unsigned long long q[4]; v8i v; } g1;
    // q0: wg_mask=0 | data_size=3 (8B) @16 | tensor_dim0[15:0] @48
    g1.q[0] = (3ull << 16) | ((td0 & 0xFFFFull) << 48);
    // q1: tensor_dim0[31:16] | tensor_dim1=1 @16 | tile_dim0 @48
    g1.q[1] = ((td0 >> 16) & 0xFFFFull) | (1ull << 16) | ((td0 & 0xFFFFull) << 48);
    // q2: tile_dim1=1 | tile_dim2=0 @16 | tensor_dim0_stride[31:0] @32
    g1.q[2] = 1ull | ((td0 & 0xFFFFFFFFull) << 32);
    // q3: tensor_dim0_stride[47:32] | tensor_dim1_stride @16
    g1.q[3] = ((td0 >> 32) & 0xFFFFull) | ((td0 & 0xFFFFFFFFFFFFull) << 16);

    union { unsigned long long q[2]; v4i v; } g2, g3;
    g2.q[0] = 0; g2.q[1] = 0; g3.q[0] = 0; g3.q[1] = 0;

    union { unsigned long long q[4]; v8i v; } g4;   // extra SGPR group (clang-23 form)
    g4.q[0] = 0; g4.q[1] = 0; g4.q[2] = 0; g4.q[3] = 0;

    __builtin_amdgcn_tensor_load_to_lds(g0.v, g1.v, g2.v, g3.v, g4.v, 0);
}
#endif

// ---------------------------------------------------------------------------
// Kernel 2: single-workgroup TDM stage -> LDS bitonic sort -> greedy NMS -> out
// LDS layout: keys[sortN]*8 | boxes[N*4]*4 | scores[N]*4 | cls[N]*4 | supp[sortN]
// (270 KB of the WGP's 320 KB for N=8192)
// ---------------------------------------------------------------------------
__global__ __launch_bounds__(1024)
void sort_nms_kernel(const unsigned long long* __restrict__ wsKeys,
                     const float* __restrict__ wsPayload,   // boxes|scores|cls blob
                     float* __restrict__ out,
                     int N, int sortN)
{
    extern __shared__ unsigned char smem[];
    unsigned long long* keys    = (unsigned long long*)smem;
    float*              lboxes  = (float*)(smem + (size_t)sortN * 8);
    float*              lscores = (float*)(smem + (size_t)sortN * 8 + (size_t)N * 16);
    float*              lcls    = (float*)(smem + (size_t)sortN * 8 + (size_t)N * 20);
    unsigned char*      supp    = (unsigned char*)(smem + (size_t)sortN * 8 + (size_t)N * 24);

    __shared__ int          sCursor, sDetCount, sGo;
    __shared__ unsigned int sDetAnchor[MAX_DET];
    __shared__ float        sPivot[4];

    const int tid  = threadIdx.x;
    const int nthr = blockDim.x;

    for (int i = tid; i < sortN; i += nthr) supp[i] = 0;
    for (int i = N + tid; i < sortN; i += nthr) keys[i] = 0ull;
    if (tid == 0) { sCursor = 0; sDetCount = 0; }

#if USE_TDM
    if ((tid >> 5) == 0) {  // wave 0 issues both TDM ops and waits on TENSORcnt
        unsigned int ldsBase = (unsigned int)(uintptr_t)(void*)smem;
        // keys: N x u64
        tdm_load_1d(ldsBase, wsKeys, (unsigned int)N);
        // payload blob (boxes+scores+cls): N*24 bytes = N*3 x u64
        tdm_load_1d(ldsBase + (unsigned int)sortN * 8u, wsPayload,
                    (unsigned int)N * 3u);
        __builtin_amdgcn_s_wait_tensorcnt(0);
    }
#else
    for (int i = tid; i < N; i += nthr) keys[i] = wsKeys[i];
    {
        unsigned long long*       ldst = (unsigned long long*)lboxes;
        const unsigned long long* lsrc = (const unsigned long long*)wsPayload;
        for (int i = tid; i < N * 3; i += nthr) ldst[i] = lsrc[i];
    }
#endif
    __syncthreads();

    // ---- bitonic sort, descending, sortN x u64 in LDS ----
    for (int k = 2; k <= sortN; k <<= 1) {
        for (int j = k >> 1; j > 0; j >>= 1) {
            for (int i = tid; i < sortN; i += nthr) {
                int ixj = i ^ j;
                if (ixj > i) {
                    bool up = ((i & k) == 0);
                    unsigned long long a = keys[i], b = keys[ixj];
                    if (up ? (a < b) : (a > b)) { keys[i] = b; keys[ixj] = a; }
                }
            }
            __syncthreads();
        }
    }

    // ---- greedy NMS with monotone cursor, early exit at MAX_DET ----
    for (int round = 0; round < MAX_DET; ++round) {
        if (tid == 0) {
            int c = sCursor;
            while (c < N && supp[c]) ++c;
            sCursor = c;
            if (c < N && keys[c] != 0ull) {
                unsigned int n = 0xFFFFFFFFu - (unsigned int)(keys[c] & 0xFFFFFFFFull);
                sDetAnchor[sDetCount] = n;
                sPivot[0] = lboxes[n * 4 + 0];
                sPivot[1] = lboxes[n * 4 + 1];
                sPivot[2] = lboxes[n * 4 + 2];
                sPivot[3] = lboxes[n * 4 + 3];
                ++sDetCount;
                supp[c] = 1;   // consumed; cursor advances past it next round
                sGo = 1;
            } else {
                sGo = 0;       // remaining entries all invalid -> done
            }
        }
        __syncthreads();
        if (!sGo) break;

        const int   c     = sCursor;
        const float px1   = sPivot[0], py1 = sPivot[1];
        const float px2   = sPivot[2], py2 = sPivot[3];
        const float parea = (px2 - px1 + 1.0f) * (py2 - py1 + 1.0f);

        for (int j = c + 1 + tid; j < N; j += nthr) {
            if (supp[j]) continue;
            unsigned long long kj = keys[j];
            if (kj == 0ull) continue;  // invalid tail
            unsigned int nj = 0xFFFFFFFFu - (unsigned int)(kj & 0xFFFFFFFFull);
            float bx1 = lboxes[nj * 4 + 0], by1 = lboxes[nj * 4 + 1];
            float bx2 = lboxes[nj * 4 + 2], by2 = lboxes[nj * 4 + 3];
            float iw = fminf(px2, bx2) - fmaxf(px1, bx1) + 1.0f;
            float ih = fminf(py2, by2) - fmaxf(py1, by1) + 1.0f;
            iw = fmaxf(iw, 0.0f);
            ih = fmaxf(ih, 0.0f);
            float inter = iw * ih;
            float barea = (bx2 - bx1 + 1.0f) * (by2 - by1 + 1.0f);
            float iou   = inter / (parea + barea - inter);
            if (iou > IOU_THRESHOLD) supp[j] = 1;
        }
        __syncthreads();
    }

    // ---- emit (MAX_DET, 7) rows: [0, x1, y1, x2, y2, score, class] ----
    for (int i = tid; i < MAX_DET * 7; i += nthr) out[i] = 0.0f;
    __syncthreads();
    if (tid < sDetCount) {
        unsigned int n = sDetAnchor[tid];
        float* row = out + tid * 7;
        row[0] = 0.0f;
        row[1] = lboxes[n * 4 + 0];
        row[2] = lboxes[n * 4 + 1];
        row[3] = lboxes[n * 4 + 2];
        row[4] = lboxes[n * 4 + 3];
        row[5] = lscores[n];
        row[6] = lcls[n];
    }
}

// ---------------------------------------------------------------------------
// Launch: inputs in setup_inputs() order: x, regression, classification, anchors
// ---------------------------------------------------------------------------
extern "C" void kernel_launch(void* const* d_in, const int* in_sizes, int n_in,
                              void* d_out, int out_size, void* d_ws, size_t ws_size,
                              hipStream_t stream)
{
    const float* reg  = (const float*)d_in[1];
    const float* cls  = (const float*)d_in[2];
    const float* anch = (const float*)d_in[3];
    float*       out  = (float*)d_out;

    const int N = in_sizes[1] / 4;          // 8192
    const int C = in_sizes[2] / N;          // 90

    // workspace layout (N*32 bytes, contiguous): keys | boxes | scores | cls
    unsigned long long* wsKeys    = (unsigned long long*)d_ws;
    float*              wsBoxes   = (float*)((char*)d_ws + (size_t)N * 8);
    float*              wsScores  = (float*)((char*)d_ws + (size_t)N * 24);
    float*              wsCls     = (float*)((char*)d_ws + (size_t)N * 28);
    const float*        wsPayload = (const float*)((char*)d_ws + (size_t)N * 8);

    int sortN = 1;
    while (sortN < N) sortN <<= 1;          // 8192 (power of two for bitonic)

    decode_score_kernel<<<(N + 255) / 256, 256, 0, stream>>>(
        reg, cls, anch, wsKeys, wsBoxes, wsScores, wsCls, N, C);

    // LDS: keys (sortN*8) + boxes/scores/cls blob (N*24) + supp (sortN)
    size_t shmem = (size_t)sortN * 8 + (size_t)N * 24 + (size_t)sortN;
    sort_nms_kernel<<<1, 1024, shmem, stream>>>(
        wsKeys, wsPayload, out, N, sortN);
}